// DepthFusionNet_79379585565311
// MI455X (gfx1250) — compile-verified
//
#include <hip/hip_runtime.h>
#include <hip/hip_bf16.h>
#include <stdint.h>
#include <stddef.h>

// ---------------------------------------------------------------------------
// MI455X (gfx1250, wave32) implementation of DepthFusionNet forward.
// Convolutions run on the fp32 matrix pipe via V_WMMA_F32_16X16X4_F32.
// - Conv geometry (KH,KW,STRIDE,PAD) templated: taps fully unrolled.
// - Weights repacked to [k-pair][co(padded16)][2]: A operand is one coalesced
//   8B load per WMMA, no predicates (zero padding baked in).
// - B operand: clamped-address loads + select (no EXEC-mask branches).
// - 1x1 stride-1 convs stage activations LDS-side with the Tensor Data Mover.
// ---------------------------------------------------------------------------

typedef float v2f __attribute__((ext_vector_type(2)));
typedef float v8f __attribute__((ext_vector_type(8)));
typedef unsigned int v4u __attribute__((ext_vector_type(4)));
typedef int v8i __attribute__((ext_vector_type(8)));
typedef int v4i __attribute__((ext_vector_type(4)));

#define DEV static __device__ __forceinline__
DEV int imin(int a, int b) { return a < b ? a : b; }
DEV int imax(int a, int b) { return a > b ? a : b; }

// ------------------------------- prep ---------------------------------------
// Per view: w2c = inv(pose)[:3] (rigid inverse), KRt = K @ w2c, Kinv analytic.
// Layout per view (36 floats): [w2c 3x4][KRt 3x4][Kinv 3x3][pad 3].
// prep[288] = -1/near, prep[289] = -1/far.
__global__ void prep_kernel(const float* __restrict__ Ks,
                            const float* __restrict__ poses,
                            const float* __restrict__ dr,
                            float* __restrict__ prep, int V) {
  int v = threadIdx.x;
  if (v == 0) {
    prep[36 * 8 + 0] = -1.0f / dr[0];
    prep[36 * 8 + 1] = -1.0f / dr[1];
  }
  if (v >= V) return;
  const float* P = poses + v * 16;
  const float* K = Ks + v * 9;
  float R[3][3], t[3];
  for (int i = 0; i < 3; ++i) {
    for (int j = 0; j < 3; ++j) R[i][j] = P[i * 4 + j];
    t[i] = P[i * 4 + 3];
  }
  float w2c[3][4];
  for (int i = 0; i < 3; ++i) {
    for (int j = 0; j < 3; ++j) w2c[i][j] = R[j][i];            // R^T
    w2c[i][3] = -(R[0][i] * t[0] + R[1][i] * t[1] + R[2][i] * t[2]);
  }
  float KRt[3][4];
  for (int i = 0; i < 3; ++i)
    for (int j = 0; j < 4; ++j) {
      float s = 0.f;
      for (int k = 0; k < 3; ++k) s += K[i * 3 + k] * w2c[k][j];
      KRt[i][j] = s;
    }
  float fx = K[0], cx = K[2], fy = K[4], cy = K[5];
  float* o = prep + v * 36;
  for (int i = 0; i < 3; ++i)
    for (int j = 0; j < 4; ++j) { o[i * 4 + j] = w2c[i][j]; o[12 + i * 4 + j] = KRt[i][j]; }
  o[24] = 1.f / fx; o[25] = 0.f;      o[26] = -cx / fx;
  o[27] = 0.f;      o[28] = 1.f / fy; o[29] = -cy / fy;
  o[30] = 0.f;      o[31] = 0.f;      o[32] = 1.f;
}

// ------------------------------- weight repack -------------------------------
// K-enumeration k' = kyx*Cin + ci.  wT layout: ((k>>1)*Coutp + co)*2 + (k&1),
// zero-padded for co >= Cout.  => A operand pairs are 8B-contiguous per lane
// and stride-8B across lanes (coalesced b64), no bounds checks in hot loop.
__global__ void repack_w_kernel(const float* __restrict__ w, float* __restrict__ wT,
                                int Cin, int KHW, int Cout, int Coutp) {
  int idx = blockIdx.x * blockDim.x + threadIdx.x;
  int K = Cin * KHW;
  if (idx >= K * Coutp) return;
  int k = idx / Coutp;
  int co = idx - k * Coutp;
  int ci = k % Cin;
  int kyx = k / Cin;
  float v = (co < Cout) ? w[(size_t)co * K + (size_t)ci * KHW + kyx] : 0.f;
  wT[((size_t)(k >> 1) * Coutp + co) * 2 + (k & 1)] = v;
}

// ------------------------------- dn / pack ----------------------------------
__global__ void dn_kernel(const float* __restrict__ depths,
                          const float* __restrict__ prep,
                          float* __restrict__ dn, float* __restrict__ dabs, int total) {
  int i = blockIdx.x * blockDim.x + threadIdx.x;
  if (i >= total) return;
  float ninv = prep[288], finv = prep[289];
  float dep = fmaxf(depths[i], 1e-5f);
  float t = (-1.f / dep - ninv) / (finv - ninv);
  t = fminf(fmaxf(t, 0.f), 1.f);
  dn[i] = t;
  dabs[i] = -1.f / (t * (finv - ninv) + ninv);
}

// grid: (HW/256, 1, V)
__global__ void pack_kernel(const float* __restrict__ imgs, const float* __restrict__ dn,
                            float* __restrict__ enc_in) {
  constexpr int HW = 256 * 256;
  const int p = blockIdx.x * blockDim.x + threadIdx.x;
  const int v = blockIdx.z;
  float* e = enc_in + (size_t)v * 12 * HW;
  e[(size_t)0 * HW + p] = imgs[(size_t)(v * 3 + 0) * HW + p];
  e[(size_t)1 * HW + p] = imgs[(size_t)(v * 3 + 1) * HW + p];
  e[(size_t)2 * HW + p] = imgs[(size_t)(v * 3 + 2) * HW + p];
  e[(size_t)3 * HW + p] = dn[(size_t)v * HW + p];
}

// ------------------------------- warp + stats -------------------------------
DEV float bilerp(const float* f, int i00, int i01, int i10, int i11, float wx, float wy) {
  float a = f[i00] * (1.f - wx) + f[i01] * wx;
  float b = f[i10] * (1.f - wx) + f[i11] * wx;
  return a * (1.f - wy) + b * wy;
}

// grid: (HW/256, 1, V); dims hardcoded (V=8, H=W=256) => shifts, no division.
__global__ void warp_stats_kernel(const float* __restrict__ imgs,
                                  const float* __restrict__ dabs,
                                  const float* __restrict__ prep,
                                  float* __restrict__ enc_in) {
  constexpr int V = 8, H = 256, W = 256, HW = H * W;
  const int rem = blockIdx.x * blockDim.x + threadIdx.x;
  const int v = blockIdx.z;
  const int y = rem >> 8;
  const int x = rem & 255;
  const float ninv = prep[288], finv = prep[289];
  const float* pv = prep + v * 36;
  const float d = dabs[(size_t)v * HW + rem];
  const float hx = (float)x, hy = (float)y;
  const float cxp = d * (pv[24] * hx + pv[25] * hy + pv[26]);
  const float cyp = d * (pv[27] * hx + pv[28] * hy + pv[29]);
  const float czp = d * (pv[30] * hx + pv[31] * hy + pv[32]);
  const float qx = cxp - pv[3], qy = cyp - pv[7], qz = czp - pv[11];
  const float X = pv[0] * qx + pv[4] * qy + pv[8] * qz;   // R^T (pcam - t)
  const float Y = pv[1] * qx + pv[5] * qy + pv[9] * qz;
  const float Z = pv[2] * qx + pv[6] * qy + pv[10] * qz;
  const float r0 = imgs[(size_t)(v * 3 + 0) * HW + rem];
  const float r1 = imgs[(size_t)(v * 3 + 1) * HW + rem];
  const float r2 = imgs[(size_t)(v * 3 + 2) * HW + rem];
  float ms = 0.f, s1[4] = {0, 0, 0, 0}, s2[4] = {0, 0, 0, 0};
  for (int u = 0; u < V; ++u) {
    const float* pu = prep + u * 36 + 12;  // KRt
    const float a0 = pu[0] * X + pu[1] * Y + pu[2] * Z + pu[3];
    const float a1 = pu[4] * X + pu[5] * Y + pu[6] * Z + pu[7];
    const float a2 = pu[8] * X + pu[9] * Y + pu[10] * Z + pu[11];
    const bool inval = fabsf(a2) < 1e-4f;
    const float dd = inval ? 1e-3f : a2;
    const float sx = a0 / dd, sy = a1 / dd;
    const bool img_inval = (sx < -0.5f) || (sx >= (float)W - 0.5f) ||
                           (sy < -0.5f) || (sy >= (float)H - 0.5f);
    const float valid = (!inval && !img_inval) ? 1.f : 0.f;
    float xc = fminf(fmaxf(sx, 0.f), (float)W - 1.f);
    float yc = fminf(fmaxf(sy, 0.f), (float)H - 1.f);
    float x0f = floorf(xc), y0f = floorf(yc);
    float wx = xc - x0f, wy = yc - y0f;
    int x0i = (int)x0f, y0i = (int)y0f;
    int x1i = (x0i + 1 < W) ? x0i + 1 : W - 1;
    int y1i = (y0i + 1 < H) ? y0i + 1 : H - 1;
    int i00 = (y0i << 8) + x0i, i01 = (y0i << 8) + x1i;
    int i10 = (y1i << 8) + x0i, i11 = (y1i << 8) + x1i;
    float f3 = bilerp(dabs + (size_t)u * HW, i00, i01, i10, i11, wx, wy);
    float feat[4];
    feat[0] = fabsf(bilerp(imgs + (size_t)(u * 3 + 0) * HW, i00, i01, i10, i11, wx, wy) - r0);
    feat[1] = fabsf(bilerp(imgs + (size_t)(u * 3 + 1) * HW, i00, i01, i10, i11, wx, wy) - r1);
    feat[2] = fabsf(bilerp(imgs + (size_t)(u * 3 + 2) * HW, i00, i01, i10, i11, wx, wy) - r2);
    float dpt_int = fmaxf(f3, 1e-5f);
    float dpt_prj = fmaxf(dd, 1e-5f);
    feat[3] = fminf(fabsf(-1.f / dpt_int + 1.f / dpt_prj) / (finv - ninv), 1.5f);
    ms += valid;
    for (int c = 0; c < 4; ++c) { s1[c] += valid * feat[c]; s2[c] += valid * feat[c] * feat[c]; }
  }
  float msc = fmaxf(ms, 1e-4f);
  float* e = enc_in + (size_t)v * 12 * HW + rem;
  for (int c = 0; c < 3; ++c) {
    float mean = s1[c] / msc;
    float var = fmaxf(s2[c] / msc - mean * mean, 0.f);
    e[(size_t)(4 + c) * HW] = mean;
    e[(size_t)(7 + c) * HW] = var;
  }
  float md = s1[3] / msc;
  e[(size_t)10 * HW] = md;
  e[(size_t)11 * HW] = fmaxf(s2[3] / msc - md * md, 0.f);
}

// ------------------------------- WMMA conv ----------------------------------
// Implicit GEMM: M = Cout tile (16), N = 16 output pixels.
// A from repacked wT (8B vector loads, no predicates); B from NCHW input via
// clamped-address loads + select.  fp32 A layout (16x4): V0 lanes0-15 -> K0,
// lanes16-31 -> K2; V1 -> K1/K3; B mirrored.
// grid: (pixTiles/4, coTiles, V); 128 threads = 4 waves, 1 tile per wave.
template <int KH, int KW, int STRIDE, int PAD>
__global__ void conv_wmma_kernel(const float* __restrict__ in, const float* __restrict__ wT,
                                 const float* __restrict__ bias, float* __restrict__ out,
                                 int Cin, int Hin, int Win,
                                 int Cout, int Coutp, int Wout, int woShift, int HWo,
                                 int outCoff, int outCtot, int flags) {
  const int lane = threadIdx.x & 31;
  const int wv = threadIdx.x >> 5;
  const int pixTile = blockIdx.x * (blockDim.x >> 5) + wv;
  const int coTile = blockIdx.y;
  const int n = blockIdx.z;
  const int half = lane >> 4, l16 = lane & 15;
  constexpr int KHW = KH * KW;
  const int pix = (pixTile << 4) + l16;
  const int oy = pix >> woShift;
  const int ox = pix & (Wout - 1);
  const int co = coTile * 16 + l16;          // < Coutp always (padded)
  const int HinWin = Hin * Win;
  const float* ibase = in + (size_t)n * Cin * HinWin;
  v8f acc = {};

  if ((Cin & 3) == 0) {
    // ---- fast path: taps unrolled, branch-free loads ----
    const int iy0 = oy * STRIDE - PAD;
    const int ix0 = ox * STRIDE - PAD;
#pragma unroll
    for (int ky = 0; ky < KH; ++ky) {
#pragma unroll
      for (int kx = 0; kx < KW; ++kx) {
        const int kyx = ky * KW + kx;
        const int iy = iy0 + ky;
        const int ix = ix0 + kx;
        const bool inb = (PAD == 0) ? true
                                    : (iy >= 0 && iy < Hin && ix >= 0 && ix < Win);
        const int iyc = (PAD == 0) ? iy : imin(imax(iy, 0), Hin - 1);
        const int ixc = (PAD == 0) ? ix : imin(imax(ix, 0), Win - 1);
        const float* ib = ibase + (size_t)iyc * Win + ixc + (size_t)(half * 2) * HinWin;
        const float* wp = wT + ((size_t)(((kyx * Cin) >> 1) + half) * Coutp + co) * 2;
#pragma unroll 4
        for (int ci = 0; ci < Cin; ci += 4) {
          v2f a = *(const v2f*)wp;              // coalesced b64
          float bx = ib[0];                     // unconditional loads
          float by = ib[(size_t)HinWin];
          v2f b;
          b.x = inb ? bx : 0.f;                 // v_cndmask, no saveexec
          b.y = inb ? by : 0.f;
          acc = __builtin_amdgcn_wmma_f32_16x16x4_f32(false, a, false, b, (short)0, acc,
                                                      false, false);
          wp += (size_t)4 * Coutp;              // pair advances by 2
          ib += (size_t)4 * HinWin;
        }
      }
    }
  } else {
    // ---- Cin == 1 path (d1 conv): k' = kyx ----
#pragma unroll
    for (int k0 = 0; k0 < KHW; k0 += 4) {
      const int ka = k0 + half * 2;
      const int kaC = imin(ka, KHW - 2);
      v2f a = *(const v2f*)&wT[((size_t)(kaC >> 1) * Coutp + co) * 2];
      v2f b;
#pragma unroll
      for (int q = 0; q < 2; ++q) {
        int k = kaC + q;
        int ky = k / KW, kx = k - (k / KW) * KW;      // constant divisors
        int iy = oy * STRIDE - PAD + ky;
        int ix = ox * STRIDE - PAD + kx;
        bool ok = (ka + q < KHW) &&
                  (PAD == 0 || (iy >= 0 && iy < Hin && ix >= 0 && ix < Win));
        int iyc = imin(imax(iy, 0), Hin - 1);
        int ixc = imin(imax(ix, 0), Win - 1);
        float vv = ibase[(size_t)iyc * Win + ixc];
        if (q == 0) b.x = ok ? vv : 0.f; else b.y = ok ? vv : 0.f;
      }
      if (ka >= KHW) { a.x = 0.f; a.y = 0.f; }
      acc = __builtin_amdgcn_wmma_f32_16x16x4_f32(false, a, false, b, (short)0, acc,
                                                  false, false);
    }
  }

#pragma unroll
  for (int i2 = 0; i2 < 8; ++i2) {
    int coO = coTile * 16 + i2 + half * 8;   // C layout: VGPR r -> M=r / M=r+8
    if (coO < Cout) {
      float v = acc[i2];
      if (flags & 1) v += bias[coO];
      if (flags & 2) v = fmaxf(v, 0.f);
      out[((size_t)n * outCtot + outCoff + coO) * HWo + pix] = v;
    }
  }
}

// --------------------- 1x1 stride-1 conv: TDM -> LDS -> WMMA -----------------
// grid: (HW/16/4, coTiles, V); weights repacked (k' = ci).
#define TDM_CIN_MAX 64
__global__ void conv1x1_tdm_kernel(const float* __restrict__ in, const float* __restrict__ wT,
                                   const float* __restrict__ bias, float* __restrict__ out,
                                   int Cin, int HW, int Cout, int Coutp,
                                   int outCoff, int outCtot, int flags) {
  __shared__ float smem[4 * TDM_CIN_MAX * 16];   // one Cin x 16 tile per wave
  const int lane = threadIdx.x & 31;
  const int wv = threadIdx.x >> 5;
  const int pixTile = blockIdx.x * (blockDim.x >> 5) + wv;
  const int coTile = blockIdx.y;
  const int n = blockIdx.z;
  const int pixBase = pixTile << 4;
  float* my = &smem[wv * TDM_CIN_MAX * 16];
  const float* gsrc = in + (size_t)n * Cin * HW + pixBase;

#if __has_builtin(__builtin_amdgcn_tensor_load_to_lds)
  {
    // Tensor DMA descriptor (CDNA5 ISA):
    // 2D tile: tile_dim0=16 elems (4B), tile_dim1=Cin rows, row stride = HW.
    // NOTE: assumes this kernel's only __shared__ array sits at LDS offset 0.
    // 6-arg (clang-23) builtin:
    //   (uint32x4 g0, int32x8 g1, int32x4 g2, int32x4 g3, int32x8 gx, i32 cpol)
    unsigned long long ga = (unsigned long long)(uintptr_t)gsrc;
    unsigned ldsb = __builtin_amdgcn_readfirstlane((unsigned)(wv * TDM_CIN_MAX * 16 * 4));
    unsigned ga_lo = __builtin_amdgcn_readfirstlane((unsigned)(ga & 0xffffffffull));
    unsigned ga_hi = __builtin_amdgcn_readfirstlane((unsigned)((ga >> 32) & 0x01ffffffull));
    v4u g0;
    g0.x = 1u;                       // count=1 (valid user descriptor)
    g0.y = ldsb;                     // lds_addr
    g0.z = ga_lo;                    // global_addr[31:0]
    g0.w = ga_hi | (2u << 30);       // global_addr[56:32] | type=2 ("image")
    unsigned td0 = 16u, td1 = (unsigned)Cin, tl0 = 16u, tl1 = (unsigned)Cin;
    unsigned long long st0 = (unsigned long long)HW;   // elements between rows
    v8i g1;
    g1[0] = (int)(2u << 16);                                       // data_size=4B, mask=0
    g1[1] = (int)((td0 & 0xffffu) << 16);                          // tensor_dim0[15:0]
    g1[2] = (int)(((td0 >> 16) & 0xffffu) | ((td1 & 0xffffu) << 16));
    g1[3] = (int)(((td1 >> 16) & 0xffffu) | (tl0 << 16));          // tile_dim0
    g1[4] = (int)(tl1 & 0xffffu);                                  // tile_dim1 (tile_dim2=0)
    g1[5] = (int)(st0 & 0xffffffffull);                            // dim0 stride lo
    g1[6] = (int)((st0 >> 32) & 0xffffu);                          // dim0 stride hi
    g1[7] = 0;
    v4i g2 = {0, 0, 0, 0}, g3 = {0, 0, 0, 0};
    v8i gx = {0, 0, 0, 0, 0, 0, 0, 0};
    __builtin_amdgcn_tensor_load_to_lds(g0, g1, g2, g3, gx, 0);
    __builtin_amdgcn_s_wait_tensorcnt(0);
  }
#else
  for (int i = lane; i < Cin * 16; i += 32) {
    int c = i >> 4, pp = i & 15;
    my[i] = gsrc[(size_t)c * HW + pp];
  }
#endif

  const int half = lane >> 4, l16 = lane & 15;
  const int co = coTile * 16 + l16;          // < Coutp always
  v8f acc = {};
#pragma unroll 4
  for (int k0 = 0; k0 < Cin; k0 += 4) {
    const int ka = k0 + half * 2;
    v2f a = *(const v2f*)&wT[((size_t)(ka >> 1) * Coutp + co) * 2];
    v2f b;
    b.x = my[ka * 16 + l16];
    b.y = my[(ka + 1) * 16 + l16];
    acc = __builtin_amdgcn_wmma_f32_16x16x4_f32(false, a, false, b, (short)0, acc, false, false);
  }
  const int pix = pixBase + l16;
#pragma unroll
  for (int i2 = 0; i2 < 8; ++i2) {
    int coO = coTile * 16 + i2 + half * 8;
    if (coO < Cout) {
      float v = acc[i2];
      if (flags & 1) v += bias[coO];
      if (flags & 2) v = fmaxf(v, 0.f);
      out[((size_t)n * outCtot + outCoff + coO) * HW + pix] = v;
    }
  }
}

// ------------------------------- instance norm ------------------------------
__global__ void inorm_kernel(float* __restrict__ x, int HW, int relu) {
  float* base = x + (size_t)blockIdx.x * HW;
  float s = 0.f, ss = 0.f;
  for (int i = threadIdx.x; i < HW; i += blockDim.x) {
    float v = base[i];
    s += v; ss += v * v;
  }
  for (int off = 16; off > 0; off >>= 1) {        // wave32 reduction
    s += __shfl_down(s, off, 32);
    ss += __shfl_down(ss, off, 32);
  }
  __shared__ float ps[16], pss[16];
  int wid = threadIdx.x >> 5;
  if ((threadIdx.x & 31) == 0) { ps[wid] = s; pss[wid] = ss; }
  __syncthreads();
  if (threadIdx.x == 0) {
    float S = 0.f, SS = 0.f;
    for (int i = 0; i < (int)(blockDim.x >> 5); ++i) { S += ps[i]; SS += pss[i]; }
    ps[0] = S; pss[0] = SS;
  }
  __syncthreads();
  float mean = ps[0] / (float)HW;
  float var = pss[0] / (float)HW - mean * mean;
  float inv = rsqrtf(var + 1e-5f);
  for (int i = threadIdx.x; i < HW; i += blockDim.x) {
    float v = (base[i] - mean) * inv;
    if (relu) v = fmaxf(v, 0.f);
    base[i] = v;
  }
}

// ------------------------------- glue ---------------------------------------
__global__ void add_relu_kernel(const float* __restrict__ a, const float* __restrict__ b,
                                float* __restrict__ o, int n) {
  int i = blockIdx.x * blockDim.x + threadIdx.x;
  if (i < n) o[i] = fmaxf(a[i] + b[i], 0.f);
}

// grid: ((Ho*Wo)/256, 1, NC); Wo is a power of two (shift/mask addressing).
__global__ void up2_kernel(const float* __restrict__ src, float* __restrict__ dst,
                           int Hin, int Win, int woShift) {
  const int nc = blockIdx.z;
  const int r = blockIdx.x * blockDim.x + threadIdx.x;
  const int Wo = Win << 1;
  const int oy = r >> woShift;
  const int ox = r & (Wo - 1);
  dst[(size_t)nc * (Hin * Win * 4) + r] =
      src[(size_t)nc * Hin * Win + (size_t)(oy >> 1) * Win + (ox >> 1)];
}

// grid: (HW/256, C, N)
__global__ void copy_ch_kernel(const float* __restrict__ src, float* __restrict__ dst,
                               int C, int HW, int dstOff, int dstCtot) {
  const int p = blockIdx.x * blockDim.x + threadIdx.x;
  const int c = blockIdx.y;
  const int n = blockIdx.z;
  dst[((size_t)n * dstCtot + dstOff + c) * HW + p] = src[((size_t)n * C + c) * HW + p];
}

// ---------------------------------------------------------------------------
extern "C" void kernel_launch(void* const* d_in, const int* in_sizes, int n_in,
                              void* d_out, int out_size, void* d_ws, size_t ws_size,
                              hipStream_t stream) {
  (void)in_sizes; (void)n_in; (void)out_size; (void)ws_size;
  // Inputs in setup_inputs() dict order; params dict flattened leaf-by-leaf.
  const float* imgs   = (const float*)d_in[0];
  /* d_in[1] = feats: unused by the reference forward pass */
  const float* depths = (const float*)d_in[2];
  const float* Ks     = (const float*)d_in[3];
  const float* poses  = (const float*)d_in[4];
  const float* dr     = (const float*)d_in[5];
  const float* c1w  = (const float*)d_in[6];
  const float* b1w1 = (const float*)d_in[7];
  const float* b1w2 = (const float*)d_in[8];
  const float* b1wd = (const float*)d_in[9];
  const float* b2w1 = (const float*)d_in[10];
  const float* b2w2 = (const float*)d_in[11];
  const float* b2wd = (const float*)d_in[12];
  const float* b3w1 = (const float*)d_in[13];
  const float* b3w2 = (const float*)d_in[14];
  const float* b3wd = (const float*)d_in[15];
  const float* u3w = (const float*)d_in[16];
  const float* u3b = (const float*)d_in[17];
  const float* i3w = (const float*)d_in[18];
  const float* i3b = (const float*)d_in[19];
  const float* u2w = (const float*)d_in[20];
  const float* u2b = (const float*)d_in[21];
  const float* i2w = (const float*)d_in[22];
  const float* i2b = (const float*)d_in[23];
  const float* oww = (const float*)d_in[24];
  const float* obb = (const float*)d_in[25];
  const float* d1w = (const float*)d_in[26];
  const float* d1b = (const float*)d_in[27];
  const float* d2w = (const float*)d_in[28];
  const float* d2b = (const float*)d_in[29];
  const float* coww = (const float*)d_in[30];
  const float* cobb = (const float*)d_in[31];

  const int V = 8, H = 256, W = 256, HW = H * W;
  float* ws = (float*)d_ws;
  size_t off = 0;
  auto alloc = [&](size_t n) { float* p = ws + off; off += n; return p; };
  // ~94 MB of fp32 scratch total (+ ~2 MB repacked weights).
  float* prep   = alloc(512);
  float* dn     = alloc((size_t)V * HW);
  float* dabs   = alloc((size_t)V * HW);
  float* enc_in = alloc((size_t)V * 12 * HW);
  float* x1     = alloc((size_t)V * 32 * 128 * 128);
  float* x2     = alloc((size_t)V * 32 * 64 * 64);
  float* x3     = alloc((size_t)V * 64 * 32 * 32);
  float* x4     = alloc((size_t)V * 128 * 16 * 16);
  float* tmpA   = alloc(1048576);
  float* tmpB   = alloc(1048576);
  float* tmpC   = alloc(1048576);
  float* cat3   = alloc((size_t)V * 128 * 32 * 32);
  float* up3    = alloc((size_t)V * 64 * 64 * 64);
  float* cat2   = alloc((size_t)V * 64 * 64 * 64);
  float* catf   = alloc((size_t)V * 48 * 64 * 64);

  auto coutp = [](int c) { return (c + 15) & ~15; };
  // repacked weights
  auto rp = [&](const float* w, int Cin, int KHW, int Cout) {
    int Cp = coutp(Cout);
    float* wT = alloc((size_t)Cin * KHW * Cp);
    int n = Cin * KHW * Cp;
    repack_w_kernel<<<(n + 255) / 256, 256, 0, stream>>>(w, wT, Cin, KHW, Cout, Cp);
    return wT;
  };
  float* c1wT  = rp(c1w, 12, 49, 32);
  float* b1w1T = rp(b1w1, 32, 9, 32);
  float* b1w2T = rp(b1w2, 32, 9, 32);
  float* b1wdT = rp(b1wd, 32, 1, 32);
  float* b2w1T = rp(b2w1, 32, 9, 64);
  float* b2w2T = rp(b2w2, 64, 9, 64);
  float* b2wdT = rp(b2wd, 32, 1, 64);
  float* b3w1T = rp(b3w1, 64, 9, 128);
  float* b3w2T = rp(b3w2, 128, 9, 128);
  float* b3wdT = rp(b3wd, 64, 1, 128);
  float* u3wT  = rp(u3w, 128, 9, 64);
  float* i3wT  = rp(i3w, 128, 9, 64);
  float* u2wT  = rp(u2w, 64, 9, 32);
  float* i2wT  = rp(i2w, 64, 9, 32);
  float* owwT  = rp(oww, 32, 1, 32);
  float* d1wT  = rp(d1w, 1, 4, 8);
  float* d2wT  = rp(d2w, 8, 4, 16);
  float* cowwT = rp(coww, 48, 1, 32);

  auto cgrid = [&](int Hout, int Wout, int Cout) {
    return dim3((unsigned)(Hout * Wout / 16 / 4), (unsigned)((Cout + 15) / 16), (unsigned)V);
  };
  auto ctz = [](int x) { int s = 0; while ((x & 1) == 0) { x >>= 1; ++s; } return s; };

#define CONV(KH, KW, S, P, inb, wg, bi, ob, Cin, Hin, Win, Cout, Hout, Wout, oCoff, oCtot, fl) \
  conv_wmma_kernel<KH, KW, S, P><<<cgrid(Hout, Wout, Cout), 128, 0, stream>>>(                 \
      inb, wg, bi, ob, Cin, Hin, Win, Cout, coutp(Cout), Wout, ctz(Wout),                      \
      (Hout) * (Wout), oCoff, oCtot, fl)

  auto conv1 = [&](const float* in, const float* wT, const float* bi, float* o,
                   int Cin, int HWp, int Cout, int oCoff, int oCtot, int flags) {
    dim3 g((unsigned)(HWp / 16 / 4), (unsigned)((Cout + 15) / 16), (unsigned)V);
    conv1x1_tdm_kernel<<<g, 128, 0, stream>>>(in, wT, bi, o, Cin, HWp, Cout, coutp(Cout),
                                              oCoff, oCtot, flags);
  };
  auto inorm = [&](float* b, int C, int HWp, int relu) {
    inorm_kernel<<<V * C, 256, 0, stream>>>(b, HWp, relu);
  };
  auto up2 = [&](const float* s, float* d, int NC, int Hin, int Win) {
    dim3 g((unsigned)(Hin * Win * 4 / 256), 1, (unsigned)NC);
    up2_kernel<<<g, 256, 0, stream>>>(s, d, Hin, Win, ctz(Win * 2));
  };
  auto copych = [&](const float* s, float* d, int C, int HWp, int dOff, int dCtot) {
    dim3 g((unsigned)(HWp / 256), (unsigned)C, (unsigned)V);
    copy_ch_kernel<<<g, 256, 0, stream>>>(s, d, C, HWp, dOff, dCtot);
  };

  // ---- geometry / warp stats ----
  prep_kernel<<<1, 32, 0, stream>>>(Ks, poses, dr, prep, V);
  int tot = V * HW;
  dn_kernel<<<(tot + 255) / 256, 256, 0, stream>>>(depths, prep, dn, dabs, tot);
  {
    dim3 g((unsigned)(HW / 256), 1, (unsigned)V);
    pack_kernel<<<g, 256, 0, stream>>>(imgs, dn, enc_in);
    warp_stats_kernel<<<g, 256, 0, stream>>>(imgs, dabs, prep, enc_in);
  }

  // ---- encoder ----
  CONV(7, 7, 2, 3, enc_in, c1wT, nullptr, x1, 12, 256, 256, 32, 128, 128, 0, 32, 0);
  inorm(x1, 32, 128 * 128, 1);
  {  // b1: (32,128,128) -> (32,64,64)
    CONV(3, 3, 2, 1, x1, b1w1T, nullptr, tmpA, 32, 128, 128, 32, 64, 64, 0, 32, 0);
    inorm(tmpA, 32, 4096, 1);
    CONV(3, 3, 1, 1, tmpA, b1w2T, nullptr, tmpB, 32, 64, 64, 32, 64, 64, 0, 32, 0);
    inorm(tmpB, 32, 4096, 0);
    CONV(1, 1, 2, 0, x1, b1wdT, nullptr, tmpC, 32, 128, 128, 32, 64, 64, 0, 32, 0);
    inorm(tmpC, 32, 4096, 0);
    int n = V * 32 * 4096;
    add_relu_kernel<<<(n + 255) / 256, 256, 0, stream>>>(tmpB, tmpC, x2, n);
  }
  {  // b2: (32,64,64) -> (64,32,32)
    CONV(3, 3, 2, 1, x2, b2w1T, nullptr, tmpA, 32, 64, 64, 64, 32, 32, 0, 64, 0);
    inorm(tmpA, 64, 1024, 1);
    CONV(3, 3, 1, 1, tmpA, b2w2T, nullptr, tmpB, 64, 32, 32, 64, 32, 32, 0, 64, 0);
    inorm(tmpB, 64, 1024, 0);
    CONV(1, 1, 2, 0, x2, b2wdT, nullptr, tmpC, 32, 64, 64, 64, 32, 32, 0, 64, 0);
    inorm(tmpC, 64, 1024, 0);
    int n = V * 64 * 1024;
    add_relu_kernel<<<(n + 255) / 256, 256, 0, stream>>>(tmpB, tmpC, x3, n);
  }
  {  // b3: (64,32,32) -> (128,16,16)
    CONV(3, 3, 2, 1, x3, b3w1T, nullptr, tmpA, 64, 32, 32, 128, 16, 16, 0, 128, 0);
    inorm(tmpA, 128, 256, 1);
    CONV(3, 3, 1, 1, tmpA, b3w2T, nullptr, tmpB, 128, 16, 16, 128, 16, 16, 0, 128, 0);
    inorm(tmpB, 128, 256, 0);
    CONV(1, 1, 2, 0, x3, b3wdT, nullptr, tmpC, 64, 32, 32, 128, 16, 16, 0, 128, 0);
    inorm(tmpC, 128, 256, 0);
    int n = V * 128 * 256;
    add_relu_kernel<<<(n + 255) / 256, 256, 0, stream>>>(tmpB, tmpC, x4, n);
  }

  // ---- decoder ----
  up2(x4, tmpA, V * 128, 16, 16);
  CONV(3, 3, 1, 1, tmpA, u3wT, u3b, cat3, 128, 32, 32, 64, 32, 32, 0, 128, 3);
  copych(x3, cat3, 64, 1024, 64, 128);
  CONV(3, 3, 1, 1, cat3, i3wT, i3b, tmpB, 128, 32, 32, 64, 32, 32, 0, 64, 3);  // y3
  up2(tmpB, up3, V * 64, 32, 32);
  CONV(3, 3, 1, 1, up3, u2wT, u2b, cat2, 64, 64, 64, 32, 64, 64, 0, 64, 3);
  copych(x2, cat2, 32, 4096, 32, 64);
  CONV(3, 3, 1, 1, cat2, i2wT, i2b, tmpA, 64, 64, 64, 32, 64, 64, 0, 32, 3);   // y2
  conv1(tmpA, owwT, obb, catf, 32, 64 * 64, 32, 16, 48, 1);                    // enc -> ch16..47

  // ---- depth branch ----
  CONV(2, 2, 2, 0, dn, d1wT, d1b, tmpC, 1, 256, 256, 8, 128, 128, 0, 8, 3);
  CONV(2, 2, 2, 0, tmpC, d2wT, d2b, catf, 8, 128, 128, 16, 64, 64, 0, 48, 1);  // ds -> ch0..15

  // ---- final 1x1 ----
  conv1(catf, cowwT, cobb, (float*)d_out, 48, 64 * 64, 32, 0, 32, 1);
#undef CONV
}